// OverlandFlowHead_41918880809700
// MI455X (gfx1250) — compile-verified
//
#include <hip/hip_runtime.h>

// ---------------------------------------------------------------------------
// Kinematic overland-flow outflow, 4096x4096 grid (MI455X / gfx1250, wave32).
// Memory-bound stencil: ~384 MB unique traffic -> ~16.5 us floor @ 23.3 TB/s.
// Threads are x-coalesced; each thread marches ROWS rows, loop-carrying the
// down-neighbor loads (mask/slopey/ptop) and the north flux. West flux comes
// from the left lane via wave32 shuffle; wave-boundary lane 0 reads the left
// wave's value from LDS (one barrier/row), so only the block's first wave
// ever recomputes. Fast 1-ulp rcp/sqrt/exp2/log2 keep VALU below the HBM
// roofline. NT store + global_prefetch_b8 for the CDNA5 memory path.
// ---------------------------------------------------------------------------

static constexpr int   NYc  = 4096;
static constexpr int   NXc  = 4096;
static constexpr int   NZc  = 5;
static constexpr int   ROWS = 8;          // rows per block (4096 % 8 == 0)
static constexpr int   BW   = 256;        // block width (threads)
static constexpr int   NW   = BW / 32;    // waves per block
static constexpr float EPSc = 1e-5f;
static constexpr float PW   = 5.0f / 3.0f;

__device__ __forceinline__ float rcp_f (float v) { return __builtin_amdgcn_rcpf(v);  }
__device__ __forceinline__ float sqrt_f(float v) { return __builtin_amdgcn_sqrtf(v); }

// nan_to_num + clamp-to-0 in one op: v_max_num_f32 quiets NaN -> returns 0.
__device__ __forceinline__ float ptop_at(const float* __restrict__ P, int y, int x) {
    return fmaxf(P[(y * NZc + (NZc - 1)) * NXc + x], 0.0f);
}

// p^(5/3) for p >= 0: v_exp_f32((5/3) * v_log_f32(p)); exact 0 for p == 0.
__device__ __forceinline__ float pow53(float p) {
    return (p > 0.0f) ? __builtin_amdgcn_exp2f(PW * __builtin_amdgcn_logf(p)) : 0.0f;
}

// Per-cell boundary-corrected slopes and 1/flux_factor (standalone form, used
// only on the rare patch-up/prologue paths).
// edge_x: mask[y,x]==0 && mask[y,x+1]!=0  (never true at x==NX-1)
// edge_y: mask[y,x]==0 && mask[y+1,x]!=0  (never true at y==NY-1)
__device__ __forceinline__ void cell_eval(const int*   __restrict__ M,
                                          const float* __restrict__ SX,
                                          const float* __restrict__ SY,
                                          const float* __restrict__ MN,
                                          int y, int x,
                                          float& sx, float& sy, float& rff) {
    const int i  = y * NXc + x;
    const int m0 = M[i];
    sx = SX[i];
    if (x + 1 < NXc && m0 == 0 && M[i + 1] != 0)   sx = SX[i + 1];
    sy = SY[i];
    if (y + 1 < NYc && m0 == 0 && M[i + NXc] != 0) sy = SY[i + NXc];
    const float slope = fmaxf(EPSc, sqrt_f(fmaf(sx, sx, sy * sy)));
    rff = rcp_f(sqrt_f(slope) * MN[i]);
}

// Full q_x at (y,x): pupwind = sx>0 ? ptop(y,x+1) : (sx<0 ? ptop(y,x) : 0)
__device__ __forceinline__ float qx_full(const float* __restrict__ P,
                                         const int*   __restrict__ M,
                                         const float* __restrict__ SX,
                                         const float* __restrict__ SY,
                                         const float* __restrict__ MN,
                                         int y, int x, float dyv) {
    float sx, sy, rff;
    cell_eval(M, SX, SY, MN, y, x, sx, sy, rff);
    const float p   = ptop_at(P, y, x);
    const float ps  = (x + 1 < NXc) ? ptop_at(P, y, x + 1) : 0.0f;
    const float pup = (sx > 0.0f) ? ps : ((sx < 0.0f) ? p : 0.0f);
    return -sx * rff * pow53(pup) * dyv;
}

// West-boundary q_x0 at (y, x==0): pupwind = sx>0 ? ptop(y,0) : 0
__device__ __forceinline__ float qx0_full(const float* __restrict__ P,
                                          const int*   __restrict__ M,
                                          const float* __restrict__ SX,
                                          const float* __restrict__ SY,
                                          const float* __restrict__ MN,
                                          int y, float dyv) {
    float sx, sy, rff;
    cell_eval(M, SX, SY, MN, y, 0, sx, sy, rff);
    const float pup = (sx > 0.0f) ? ptop_at(P, y, 0) : 0.0f;
    return -sx * rff * pow53(pup) * dyv;
}

// Full q_y at (y,x): pupwind = sy>0 ? ptop(y+1,x) : (sy<0 ? ptop(y,x) : 0)
__device__ __forceinline__ float qy_full(const float* __restrict__ P,
                                         const int*   __restrict__ M,
                                         const float* __restrict__ SX,
                                         const float* __restrict__ SY,
                                         const float* __restrict__ MN,
                                         int y, int x, float dxv) {
    float sx, sy, rff;
    cell_eval(M, SX, SY, MN, y, x, sx, sy, rff);
    const float p   = ptop_at(P, y, x);
    const float ps  = (y + 1 < NYc) ? ptop_at(P, y + 1, x) : 0.0f;
    const float pup = (sy > 0.0f) ? ps : ((sy < 0.0f) ? p : 0.0f);
    return -sy * rff * pow53(pup) * dxv;
}

// North-boundary q_y0 at (y==0, x): pupwind = sy>0 ? ptop(0,x) : 0
__device__ __forceinline__ float qy0_full(const float* __restrict__ P,
                                          const int*   __restrict__ M,
                                          const float* __restrict__ SX,
                                          const float* __restrict__ SY,
                                          const float* __restrict__ MN,
                                          int x, float dxv) {
    float sx, sy, rff;
    cell_eval(M, SX, SY, MN, 0, x, sx, sy, rff);
    const float pup = (sy > 0.0f) ? ptop_at(P, 0, x) : 0.0f;
    return -sy * rff * pow53(pup) * dxv;
}

__global__ __launch_bounds__(BW)
void OverlandFlowHead_kernel(const float* __restrict__ P,
                             const float* __restrict__ SX,
                             const float* __restrict__ SY,
                             const float* __restrict__ MN,
                             const float* __restrict__ dxp,
                             const float* __restrict__ dyp,
                             const int*   __restrict__ M,
                             float*       __restrict__ out) {
    __shared__ float lds_qx[2][NW];       // lane-31 q_x per wave, row-parity buffered

    const int x    = blockIdx.x * BW + threadIdx.x;   // 0..NX-1, coalesced
    const int y0   = blockIdx.y * ROWS;
    const int lane = threadIdx.x & 31;                // wave32
    const int wv   = threadIdx.x >> 5;
    const float dxv = dxp[0];
    const float dyv = dyp[0];

    const bool have_rt = (x + 1 < NXc);               // false only for global last lane
    const int  xr      = have_rt ? x + 1 : x;         // clamped right index

    // ---- prologue: loop-carried row-y0 values + north flux entering y0 ----
    const int i0 = y0 * NXc + x;
    int   m_cur  = M[i0];
    float sy_cur = SY[i0];
    float p_cur  = ptop_at(P, y0, x);
    float qy_prev = (y0 == 0) ? qy0_full(P, M, SX, SY, MN, x, dxv)
                              : qy_full (P, M, SX, SY, MN, y0 - 1, x, dxv);

    for (int y = y0; y < y0 + ROWS; ++y) {
        const int  i       = y * NXc + x;
        const bool have_dn = (y + 1 < NYc);           // uniform per row
        const int  yd      = have_dn ? y + 1 : y;     // clamped down index

        // ---- this row's loads (down-neighbors feed the next iteration) ----
        const int   m_rt  = M [y  * NXc + xr];
        const int   m_dn  = M [yd * NXc + x ];
        const float sx_c  = SX[i];
        const float sx_rt = SX[y  * NXc + xr];
        const float sy_dn = SY[yd * NXc + x ];
        const float p_rt  = ptop_at(P, y,  xr);
        const float p_dn  = ptop_at(P, yd, x );
        const float mn    = MN[i];

        // Prefetch pressure top-slice two rows ahead (global_prefetch_b8).
        if (y + 2 < NYc)
            __builtin_prefetch(&P[((y + 2) * NZc + (NZc - 1)) * NXc + x], 0, 1);

        // ---- boundary-corrected slopes, shared flux factor ----
        const bool edge_x = have_rt && (m_cur == 0) && (m_rt != 0);
        const bool edge_y = have_dn && (m_cur == 0) && (m_dn != 0);
        const float sx = edge_x ? sx_rt : sx_c;
        const float sy = edge_y ? sy_dn : sy_cur;
        const float slope = fmaxf(EPSc, sqrt_f(fmaf(sx, sx, sy * sy)));
        const float rff   = rcp_f(sqrt_f(slope) * mn);

        // ---- upwind pressures and fluxes ----
        const float psx  = have_rt ? p_rt : 0.0f;
        const float psy  = have_dn ? p_dn : 0.0f;
        const float pupx = (sx > 0.0f) ? psx : ((sx < 0.0f) ? p_cur : 0.0f);
        const float pupy = (sy > 0.0f) ? psy : ((sy < 0.0f) ? p_cur : 0.0f);
        const float qx   = -sx * rff * pow53(pupx) * dyv;
        const float qy   = -sy * rff * pow53(pupy) * dxv;

        // ---- west flux: left lane via shuffle; wave seams via LDS ----
        float qx_w = __shfl_up(qx, 1);
        if (lane == 31) lds_qx[y & 1][wv] = qx;
        __syncthreads();
        if (lane == 0) {
            if (wv > 0) {
                qx_w = lds_qx[y & 1][wv - 1];
            } else {
                qx_w = (x == 0) ? qx0_full(P, M, SX, SY, MN, y, dyv)
                                : qx_full (P, M, SX, SY, MN, y, x - 1, dyv);
            }
        }

        // ---- combine, mask, non-temporal store ----
        float o = fmaxf(qx, 0.0f) + fmaxf(-qx_w, 0.0f)
                + fmaxf(qy, 0.0f) + fmaxf(-qy_prev, 0.0f);
        if (m_cur == 0) o = 0.0f;
        __builtin_nontemporal_store(o, out + i);

        // ---- carry down-neighbors into the next row ----
        qy_prev = qy;
        m_cur   = m_dn;
        sy_cur  = sy_dn;
        p_cur   = p_dn;
    }
}

extern "C" void kernel_launch(void* const* d_in, const int* in_sizes, int n_in,
                              void* d_out, int out_size, void* d_ws, size_t ws_size,
                              hipStream_t stream) {
    // setup_inputs() order: pressure, slopex, slopey, mannings, dx, dy, mask
    const float* P   = (const float*)d_in[0];
    const float* SX  = (const float*)d_in[1];
    const float* SY  = (const float*)d_in[2];
    const float* MN  = (const float*)d_in[3];
    const float* dxp = (const float*)d_in[4];
    const float* dyp = (const float*)d_in[5];
    const int*   M   = (const int*)  d_in[6];
    float* out = (float*)d_out;

    dim3 block(BW, 1, 1);
    dim3 grid(NXc / BW, NYc / ROWS, 1);
    OverlandFlowHead_kernel<<<grid, block, 0, stream>>>(P, SX, SY, MN, dxp, dyp, M, out);
}